// GRUModel_41644002902358
// MI455X (gfx1250) — compile-verified
//
#include <hip/hip_runtime.h>
#include <hip/hip_bf16.h>

typedef __attribute__((ext_vector_type(2))) float v2f;
typedef __attribute__((ext_vector_type(8))) float v8f;

#define NROWS 262144
#define PP 64
#define QQ 64
#define TSEG 2048
#define JJ 512
#define BB 8
#define HH 64

#define NEG_INF (-3.402823466e38f)

// ---------------------------------------------------------------------------
// Async global->LDS helpers (CDNA5 ASYNCcnt path, ISA ch.10 / 15.18)
// ---------------------------------------------------------------------------
__device__ __forceinline__ void async_copy_b128(void* lds_dst, const void* gsrc) {
    // VDST vgpr holds the LDS byte offset; VADDR pair holds the 64-bit address.
    unsigned ldsoff = (unsigned)(unsigned long long)lds_dst;  // low 32 bits = LDS offset
    asm volatile("global_load_async_to_lds_b128 %0, %1, off"
                 :: "v"(ldsoff), "v"((unsigned long long)gsrc)
                 : "memory");
}
__device__ __forceinline__ void wait_async_le8() {
    asm volatile("s_wait_asynccnt 0x8" ::: "memory");
}
__device__ __forceinline__ void wait_async_0() {
    asm volatile("s_wait_asynccnt 0x0" ::: "memory");
}

// ---------------------------------------------------------------------------
// K0: gather selected S columns: Ssel[p*8+b] = S[p*512 + idx[b]]
// ---------------------------------------------------------------------------
__global__ void gather_s_kernel(const float* __restrict__ S,
                                const int* __restrict__ idx,
                                float* __restrict__ Ssel) {
    int tid = threadIdx.x;            // 512 threads
    int p = tid >> 3;
    int b = tid & 7;
    Ssel[tid] = S[(size_t)p * JJ + idx[b]];
}

// ---------------------------------------------------------------------------
// K1: segment bounds via binary search over sorted seg_ids.
// seg_start[t] = first u with seg_ids[u] >= t ; seg_start[T] = N
// ---------------------------------------------------------------------------
__global__ void seg_bounds_kernel(const int* __restrict__ seg_ids,
                                  int* __restrict__ seg_start) {
    int t = blockIdx.x * blockDim.x + threadIdx.x;
    if (t > TSEG) return;
    int lo = 0, hi = NROWS;
    while (lo < hi) {
        int mid = (lo + hi) >> 1;
        if (seg_ids[mid] < t) lo = mid + 1; else hi = mid;
    }
    seg_start[t] = lo;
}

// ---------------------------------------------------------------------------
// K2: per-segment fused  w = K@Ssel,  softmax-max,  M = (sum e*V)/(sum e)
// One 256-thread block (8 waves) per segment.  Segments streamed in 128-row
// chunks through a double-buffered LDS pipeline fed by async global->LDS
// copies (ASYNCcnt), with w computed via V_WMMA_F32_16X16X4_F32.
// ---------------------------------------------------------------------------
__global__ __launch_bounds__(256) void seg_softmax_kernel(
    const float* __restrict__ K, const float* __restrict__ V,
    const float* __restrict__ Ssel, const int* __restrict__ seg_start,
    float* __restrict__ w_buf, float* __restrict__ Mbuf) {

    __shared__ float dbuf[2][128 * 64];   // double-buffered K/V chunks (64KB)
    __shared__ float e_sh[128 * 8];
    __shared__ float ssel_sh[64 * 16];    // B-matrix, cols 8..15 zero-padded
    __shared__ float wavemax_sh[8 * 8];
    __shared__ float segmax_sh[8];

    const int tid = threadIdx.x;
    const int t = blockIdx.x;
    const int s0 = seg_start[t];
    const int s1 = seg_start[t + 1];
    const int len = s1 - s0;

    if (len <= 0) {  // empty segment -> reference yields zeros
        for (int p = tid; p < BB * QQ; p += 256)
            Mbuf[(size_t)t * (BB * QQ) + p] = 0.f;
        return;
    }

    for (int i = tid; i < 64 * 16; i += 256) {
        int c = i & 15;
        ssel_sh[i] = (c < 8) ? Ssel[(i >> 4) * 8 + c] : 0.f;
    }
    if (tid < 8) segmax_sh[tid] = NEG_INF;

    const int wid  = tid >> 5;
    const int lane = tid & 31;
    const int half = lane >> 4;   // 0: lanes 0-15, 1: lanes 16-31
    const int lrow = lane & 15;
    const int nchunks = (len + 127) >> 7;

    // issue exactly 8 async b128 copies per lane (clamped rows keep the
    // per-wave issue count fixed so s_wait_asynccnt bookkeeping is exact)
    auto issue_chunk = [&](const float* __restrict__ src, int base, int cnt,
                           int bufidx) {
#pragma unroll
        for (int i = 0; i < 8; ++i) {
            int idx4 = tid + i * 256;
            int row = idx4 >> 4, c4 = idx4 & 15;
            int rc = (row < cnt) ? row : cnt - 1;
            async_copy_b128(&dbuf[bufidx][row * 64 + c4 * 4],
                            src + (size_t)(base + rc) * 64 + c4 * 4);
        }
    };

    // ---------------- Phase A: w = K @ Ssel, running per-b max -------------
    {
        int cnt0 = (len < 128) ? len : 128;
        issue_chunk(K, s0, cnt0, 0);
    }
    for (int ci = 0; ci < nchunks; ++ci) {
        const int base = s0 + ci * 128;
        int cnt = s1 - base; if (cnt > 128) cnt = 128;
        const int cur = ci & 1;

        if (ci + 1 < nchunks) {        // overlap next chunk's DMA with compute
            int nb = base + 128;
            int ncnt = s1 - nb; if (ncnt > 128) ncnt = 128;
            issue_chunk(K, nb, ncnt, cur ^ 1);
            wait_async_le8();          // current chunk has landed
        } else {
            wait_async_0();
        }
        if (cnt < 128) {               // zero-pad tail rows for the WMMA
#pragma unroll
            for (int i = 0; i < 8; ++i) {
                int idx4 = tid + i * 256;
                int row = idx4 >> 4, c4 = idx4 & 15;
                if (row >= cnt)
                    *reinterpret_cast<float4*>(&dbuf[cur][row * 64 + c4 * 4]) =
                        make_float4(0.f, 0.f, 0.f, 0.f);
            }
        }
        __syncthreads();

        const float* __restrict__ ksh = dbuf[cur];
        const int rbase = wid * 16;    // wave-uniform
        float lm = NEG_INF;
        if (rbase < cnt) {             // wave-uniform: EXEC all ones inside
            v8f c = {};
#pragma unroll
            for (int kk = 0; kk < 16; ++kk) {
                int cc = kk * 4 + half * 2;
                v2f a, b;
                a.x = ksh[(rbase + lrow) * 64 + cc];
                a.y = ksh[(rbase + lrow) * 64 + cc + 1];
                b.x = ssel_sh[(cc + 0) * 16 + lrow];
                b.y = ssel_sh[(cc + 1) * 16 + lrow];
                c = __builtin_amdgcn_wmma_f32_16x16x4_f32(
                        false, a, false, b, (short)0, c, false, false);
            }
            if (lrow < 8) {            // only cols 0..7 are real b-indices
#pragma unroll
                for (int v = 0; v < 8; ++v) {
                    int m = v + half * 8;
                    if (rbase + m < cnt) {
                        float wv = c[v];
                        w_buf[(size_t)(base + rbase + m) * BB + lrow] = wv;
                        lm = fmaxf(lm, wv);
                    }
                }
            }
        }
        lm = fmaxf(lm, __shfl_xor(lm, 16, 32));  // combine the two row-halves
        if (lane < 8) wavemax_sh[wid * 8 + lane] = lm;
        __syncthreads();
        if (tid < 8) {
            float mm = segmax_sh[tid];
#pragma unroll
            for (int w = 0; w < 8; ++w) mm = fmaxf(mm, wavemax_sh[w * 8 + tid]);
            segmax_sh[tid] = mm;
        }
        __syncthreads();   // also releases dbuf[cur] for reuse at ci+2
    }

    // ---------------- Phase B: e = exp(w-m); M = (sum e*V) / (sum e) -------
    const int b0 = tid >> 6;     // 0..3   (pairs with b0+4)
    const int q0 = tid & 63;
    float acc0 = 0.f, acc1 = 0.f, dl0 = 0.f, dl1 = 0.f;

    {
        int cnt0 = (len < 128) ? len : 128;
        issue_chunk(V, s0, cnt0, 0);
    }
    for (int ci = 0; ci < nchunks; ++ci) {
        const int base = s0 + ci * 128;
        int cnt = s1 - base; if (cnt > 128) cnt = 128;
        const int cur = ci & 1;

        if (ci + 1 < nchunks) {
            int nb = base + 128;
            int ncnt = s1 - nb; if (ncnt > 128) ncnt = 128;
            issue_chunk(V, nb, ncnt, cur ^ 1);
            // prefetch next chunk's logits while the DMA runs
            __builtin_prefetch(w_buf + (size_t)nb * BB + tid * 4, 0, 0);
            wait_async_le8();
        } else {
            wait_async_0();
        }
        // e for the current chunk (regular loads + VALU exp)
#pragma unroll
        for (int i = 0; i < 4; ++i) {
            int p = tid + i * 256;         // p = row*8 + b
            int row = p >> 3, b = p & 7;
            float e = 0.f;
            if (row < cnt)
                e = __expf(w_buf[(size_t)(base + row) * BB + b] - segmax_sh[b]);
            e_sh[p] = e;
        }
        __syncthreads();

        const float* __restrict__ vsh = dbuf[cur];
        for (int r = 0; r < cnt; ++r) {
            float e0 = e_sh[r * 8 + b0];
            float e1 = e_sh[r * 8 + b0 + 4];
            float vv = vsh[r * 64 + q0];
            acc0 += e0 * vv;
            acc1 += e1 * vv;
            dl0 += e0;
            dl1 += e1;
        }
        __syncthreads();   // e_sh + dbuf[cur] free for next iteration
    }
    // dl is the full segment denominator (identical across threads sharing b)
    Mbuf[(size_t)t * (BB * QQ) + b0 * QQ + q0]       = acc0 / dl0;
    Mbuf[(size_t)t * (BB * QQ) + (b0 + 4) * QQ + q0] = acc1 / dl1;
}

// ---------------------------------------------------------------------------
// K3: GRU over 8 time steps, batch T=2048, H=Q=64, with f32 WMMA gemms.
// 16 blocks x 8 waves; each wave owns 16 batch rows. h kept as four 16x16
// f32 accumulator fragments + LDS mirror (re-read in A-fragment layout).
// ---------------------------------------------------------------------------
__device__ __forceinline__ v8f splat8(float x) {
    v8f r;
#pragma unroll
    for (int v = 0; v < 8; ++v) r[v] = x;
    return r;
}
__device__ __forceinline__ v8f sigmoid8(v8f x) {
    v8f r;
#pragma unroll
    for (int v = 0; v < 8; ++v) r[v] = 1.f / (1.f + __expf(-x[v]));
    return r;
}

__global__ __launch_bounds__(256) void gru_kernel(
    const float* __restrict__ Mbuf,
    const float* __restrict__ W_ih, const float* __restrict__ W_hh,
    const float* __restrict__ b_ih, const float* __restrict__ b_hh,
    const float* __restrict__ w_out, const float* __restrict__ b_out,
    float* __restrict__ preds) {

    __shared__ float h_sh[8 * 16 * 64];   // per-wave h mirror (32KB)

    const int tid = threadIdx.x;
    const int wid = tid >> 5;
    const int lane = tid & 31;
    const int half = lane >> 4;
    const int lrow = lane & 15;
    const int t0 = (blockIdx.x * 8 + wid) * 16;   // wave's batch-row base
    float* hrow_base = h_sh + wid * (16 * 64);

    for (int i = lane; i < 16 * 64; i += 32) hrow_base[i] = 0.f;
    v8f h[4];
#pragma unroll
    for (int j = 0; j < 4; ++j) h[j] = splat8(0.f);
    __syncthreads();

    const float bout = b_out[0];

    // one 16x16 tile of X @ W^T accumulated over K=64 (16 wmma k-steps);
    // B fragment = W^T read directly (row-major W[g][q], g = tile*16+col)
    auto tile_gemm = [&](int tileIdx, const float* __restrict__ W,
                         const v2f* A, v8f acc) -> v8f {
        const float* wr = W + (size_t)(tileIdx * 16 + lrow) * 64;
#pragma unroll
        for (int kk = 0; kk < 16; ++kk) {
            int cc = kk * 4 + half * 2;
            v2f b;
            b.x = wr[cc];
            b.y = wr[cc + 1];
            acc = __builtin_amdgcn_wmma_f32_16x16x4_f32(
                    false, A[kk], false, b, (short)0, acc, false, false);
        }
        return acc;
    };

    for (int s = 0; s < 8; ++s) {
        // A-fragments for x_t (from Mbuf[t][s][q]) and h (from LDS mirror)
        v2f xa[16], ha[16];
        const float* xrow = Mbuf + (size_t)(t0 + lrow) * (BB * QQ) + s * QQ;
        const float* hrow = hrow_base + lrow * 64;
#pragma unroll
        for (int kk = 0; kk < 16; ++kk) {
            int cc = kk * 4 + half * 2;
            xa[kk].x = xrow[cc];  xa[kk].y = xrow[cc + 1];
            ha[kk].x = hrow[cc];  ha[kk].y = hrow[cc + 1];
        }

        v8f zg[4];
#pragma unroll
        for (int j = 0; j < 4; ++j) {
            int g = j * 16 + lrow;
            // r gate (tiles 0..3): gi_r + gh_r + both biases
            v8f accr = splat8(b_ih[g] + b_hh[g]);
            accr = tile_gemm(j, W_ih, xa, accr);
            accr = tile_gemm(j, W_hh, ha, accr);
            v8f rg = sigmoid8(accr);
            // z gate (tiles 4..7)
            v8f accz = splat8(b_ih[HH + g] + b_hh[HH + g]);
            accz = tile_gemm(4 + j, W_ih, xa, accz);
            accz = tile_gemm(4 + j, W_hh, ha, accz);
            zg[j] = sigmoid8(accz);
            // n gate (tiles 8..11): needs i_n and h_n separately
            v8f ai = splat8(b_ih[2 * HH + g]);
            ai = tile_gemm(8 + j, W_ih, xa, ai);
            v8f ah = splat8(b_hh[2 * HH + g]);
            ah = tile_gemm(8 + j, W_hh, ha, ah);
            v8f hn;
#pragma unroll
            for (int v = 0; v < 8; ++v) hn[v] = tanhf(ai[v] + rg[v] * ah[v]);
            // h' = (1-z)*n + z*h
#pragma unroll
            for (int v = 0; v < 8; ++v)
                h[j][v] = (1.f - zg[j][v]) * hn[v] + zg[j][v] * h[j][v];
        }

        __syncthreads();
        // mirror h' to LDS (C-fragment -> row-major): lane holds col lrow of
        // tile j, rows v + half*8
#pragma unroll
        for (int j = 0; j < 4; ++j)
#pragma unroll
            for (int v = 0; v < 8; ++v)
                hrow_base[(v + half * 8) * 64 + j * 16 + lrow] = h[j][v];
        __syncthreads();

        // output projection: preds[s, t] = h' . w_out + b_out
        if (lane < 16) {
            const float* hr = hrow_base + lane * 64;
            float sum = bout;
#pragma unroll
            for (int q = 0; q < 64; ++q) sum += hr[q] * w_out[q];
            preds[s * TSEG + t0 + lane] = sum;
        }
        __syncthreads();
    }
}

// ---------------------------------------------------------------------------
// Launcher
// ---------------------------------------------------------------------------
extern "C" void kernel_launch(void* const* d_in, const int* in_sizes, int n_in,
                              void* d_out, int out_size, void* d_ws, size_t ws_size,
                              hipStream_t stream) {
    const int*   idx     = (const int*)d_in[0];
    const int*   seg_ids = (const int*)d_in[1];
    const float* K       = (const float*)d_in[2];
    const float* V       = (const float*)d_in[3];
    const float* S       = (const float*)d_in[4];
    const float* W_ih    = (const float*)d_in[5];
    const float* W_hh    = (const float*)d_in[6];
    const float* b_ih    = (const float*)d_in[7];
    const float* b_hh    = (const float*)d_in[8];
    const float* w_out   = (const float*)d_in[9];
    const float* b_out   = (const float*)d_in[10];
    float* out = (float*)d_out;

    char* ws = (char*)d_ws;
    size_t off = 0;
    float* w_buf = (float*)(ws + off); off += (size_t)NROWS * BB * sizeof(float);     // 8 MB
    float* Mbuf  = (float*)(ws + off); off += (size_t)TSEG * BB * QQ * sizeof(float); // 4 MB
    float* Ssel  = (float*)(ws + off); off += (size_t)PP * BB * sizeof(float);
    int* seg_start = (int*)(ws + off); off += (size_t)(TSEG + 1) * sizeof(int);

    gather_s_kernel<<<1, 512, 0, stream>>>(S, idx, Ssel);
    seg_bounds_kernel<<<(TSEG + 1 + 255) / 256, 256, 0, stream>>>(seg_ids, seg_start);
    seg_softmax_kernel<<<TSEG, 256, 0, stream>>>(K, V, Ssel, seg_start, w_buf, Mbuf);
    gru_kernel<<<16, 256, 0, stream>>>(Mbuf, W_ih, W_hh, b_ih, b_hh, w_out, b_out, out);
}